// gat_seq_36240934044135
// MI455X (gfx1250) — compile-verified
//
#include <hip/hip_runtime.h>
#include <hip/hip_bf16.h>

#define N_NODES 100000
#define N_EDGES 1600000
#define IN_CH   128
#define EDGE_CH 32
#define HEADS   4
#define OUT_CH  32
#define HC      128            // HEADS*OUT_CH
#define NEG_SLOPE 0.2f
#define EPS 1e-16f

typedef float v2f __attribute__((ext_vector_type(2)));
typedef float v8f __attribute__((ext_vector_type(8)));

// ---- monotonic float<->uint encoding so atomicMax(uint) == float max ----
__device__ __forceinline__ unsigned encF(float f) {
    unsigned u = __float_as_uint(f);
    return u ^ ((unsigned)((int)u >> 31) | 0x80000000u);
}
__device__ __forceinline__ float decF(unsigned u) {
    unsigned v = u ^ (((u & 0x80000000u) ? 0x80000000u : 0xFFFFFFFFu));
    return __uint_as_float(v);
}

// ---------------------------------------------------------------------------
// K0: out[n*128+c] = bias[c]; amaxU = 0 (< any encoded finite float); denom = 0
// ---------------------------------------------------------------------------
__global__ void gat_init_kernel(float* __restrict__ out, const float* __restrict__ bias,
                                unsigned* __restrict__ amaxU, float* __restrict__ denom) {
    const long long nOut = (long long)N_NODES * HC;
    const int nSeg = N_NODES * HEADS;
    long long stride = (long long)gridDim.x * blockDim.x;
    for (long long i = blockIdx.x * (long long)blockDim.x + threadIdx.x; i < nOut; i += stride)
        out[i] = bias[(int)(i & (HC - 1))];
    for (long long i = blockIdx.x * (long long)blockDim.x + threadIdx.x; i < nSeg; i += stride) {
        amaxU[i] = 0u;
        denom[i] = 0.0f;
    }
}

// ---------------------------------------------------------------------------
// K1: xl[N,128] = x[N,128] @ Wl[128,128]^T via V_WMMA_F32_16X16X4_F32.
// Block = 256 thr = 8 waves; block b owns rows [16b,16b+16); wave w owns cols
// [16w,16w+16). A frag (16x4 f32): lane m=lane&15, K-half=(lane>=16)?2:0 ->
// contiguous float2 from x. B frag (4x16): B[k,n]=Wl[col_n,k] -> contiguous
// float2 from Wl row. 100000 = 16*6250 exactly (no M tail).
// ---------------------------------------------------------------------------
__global__ __launch_bounds__(256) void gat_gemm_xl_kernel(const float* __restrict__ x,
                                                          const float* __restrict__ Wl,
                                                          float* __restrict__ xl) {
    const int lane  = threadIdx.x & 31;
    const int wave  = threadIdx.x >> 5;          // N-tile 0..7
    const int rowB  = blockIdx.x << 4;           // M block base
    const int m     = lane & 15;
    const int khalf = (lane >> 4) << 1;          // 0 or 2

    const float* __restrict__ xrow = x  + (long long)(rowB + m) * IN_CH;
    const float* __restrict__ wrow = Wl + (long long)(wave * 16 + m) * IN_CH;

    v8f acc = {0.f, 0.f, 0.f, 0.f, 0.f, 0.f, 0.f, 0.f};
    #pragma unroll 8
    for (int k = 0; k < IN_CH; k += 4) {
        v2f a = *(const v2f*)(xrow + k + khalf);
        v2f b = *(const v2f*)(wrow + k + khalf);
        acc = __builtin_amdgcn_wmma_f32_16x16x4_f32(false, a, false, b,
                                                    (short)0, acc, false, false);
    }
    // D layout: VGPR r -> M = r (+8 for lanes 16-31), N = lane&15
    const int mo = rowB + ((lane >> 4) << 3);
    const int col = wave * 16 + m;
    #pragma unroll
    for (int r = 0; r < 8; ++r)
        xl[(long long)(mo + r) * HC + col] = acc[r];
}

// ---------------------------------------------------------------------------
// K2: alpha_l[n,h], alpha_r[n,h] = dot(xl[n, h*32:(h+1)*32], att_{l,r}[h,:])
// ---------------------------------------------------------------------------
__global__ void gat_alpha_lr_kernel(const float* __restrict__ xl,
                                    const float* __restrict__ att_l,
                                    const float* __restrict__ att_r,
                                    float* __restrict__ alpha_l,
                                    float* __restrict__ alpha_r) {
    int t = blockIdx.x * blockDim.x + threadIdx.x;
    if (t >= N_NODES * HEADS) return;
    int h = t & (HEADS - 1);
    int n = t >> 2;
    const float* p  = xl + (long long)n * HC + h * OUT_CH;
    const float* al = att_l + h * OUT_CH;
    const float* ar = att_r + h * OUT_CH;
    float sl = 0.f, sr = 0.f;
    #pragma unroll
    for (int c = 0; c < OUT_CH; ++c) {
        float v = p[c];
        sl = fmaf(v, al[c], sl);
        sr = fmaf(v, ar[c], sr);
    }
    alpha_l[t] = sl;
    alpha_r[t] = sr;
}

// ---------------------------------------------------------------------------
// K3: ve[k*4+h] = sum_c We[(h*32+c)*32 + k] * att_e[h*32+c]   (32x4 = 128 vals)
// Collapses the 1.6M x 128 edge projection to a 32x4 matvec per edge.
// ---------------------------------------------------------------------------
__global__ void gat_ve_kernel(const float* __restrict__ We,
                              const float* __restrict__ att_e,
                              float* __restrict__ ve) {
    int t = threadIdx.x;                 // 0..127, single block
    int k = t >> 2;
    int h = t & 3;
    float s = 0.f;
    #pragma unroll
    for (int c = 0; c < OUT_CH; ++c)
        s = fmaf(We[(h * OUT_CH + c) * EDGE_CH + k], att_e[h * OUT_CH + c], s);
    ve[t] = s;
}

// ---------------------------------------------------------------------------
// K4: per edge: alpha[e,h] = leaky_relu(alpha_l[src]+alpha_r[dst]+edge_attr@ve)
//     and encoded atomicMax into amaxU[dst,h]  (segment max)
// ---------------------------------------------------------------------------
__global__ __launch_bounds__(256) void gat_logit_kernel(const float* __restrict__ edge_attr,
                                                        const float* __restrict__ ve,
                                                        const float* __restrict__ alpha_l,
                                                        const float* __restrict__ alpha_r,
                                                        const int* __restrict__ ei,
                                                        float* __restrict__ alpha,
                                                        unsigned* __restrict__ amaxU) {
    __shared__ float sve[EDGE_CH * HEADS];
    if (threadIdx.x < EDGE_CH * HEADS) sve[threadIdx.x] = ve[threadIdx.x];
    __syncthreads();

    int e = blockIdx.x * blockDim.x + threadIdx.x;
    if (e >= N_EDGES) return;
    int src = ei[e];
    int dst = ei[N_EDGES + e];

    const float* ea = edge_attr + (long long)e * EDGE_CH;
    float s[HEADS] = {0.f, 0.f, 0.f, 0.f};
    #pragma unroll
    for (int k = 0; k < EDGE_CH; ++k) {
        float v = ea[k];
        s[0] = fmaf(v, sve[k * 4 + 0], s[0]);
        s[1] = fmaf(v, sve[k * 4 + 1], s[1]);
        s[2] = fmaf(v, sve[k * 4 + 2], s[2]);
        s[3] = fmaf(v, sve[k * 4 + 3], s[3]);
    }
    const float4 al = *(const float4*)(alpha_l + src * 4);
    const float4 ar = *(const float4*)(alpha_r + dst * 4);
    float lg[HEADS] = {s[0] + al.x + ar.x, s[1] + al.y + ar.y,
                       s[2] + al.z + ar.z, s[3] + al.w + ar.w};
    #pragma unroll
    for (int h = 0; h < HEADS; ++h) {
        float a = lg[h] > 0.f ? lg[h] : NEG_SLOPE * lg[h];
        alpha[e * 4 + h] = a;
        atomicMax(&amaxU[dst * 4 + h], encF(a));
    }
}

// ---------------------------------------------------------------------------
// K5: ex = exp(alpha - amax[dst]); overwrite alpha; atomicAdd denom[dst,h]
// ---------------------------------------------------------------------------
__global__ __launch_bounds__(256) void gat_exp_kernel(float* __restrict__ alpha,
                                                      const unsigned* __restrict__ amaxU,
                                                      const int* __restrict__ ei,
                                                      float* __restrict__ denom) {
    int t = blockIdx.x * blockDim.x + threadIdx.x;
    if (t >= N_EDGES * HEADS) return;
    int e = t >> 2;
    int h = t & 3;
    int dst = ei[N_EDGES + e];
    float mx = decF(amaxU[dst * 4 + h]);
    float ex = __expf(alpha[t] - mx);
    alpha[t] = ex;
    atomicAdd(&denom[dst * 4 + h], ex);
}

// ---------------------------------------------------------------------------
// K6: one wave per edge; lane l owns channels [4l,4l+4), head = l/8.
//     out[dst] += xl[src] * ex/(denom[dst]+EPS)   (float atomics; xl & out
//     both resident in the 192MB L2, so gather+atomic stays on-chip)
// ---------------------------------------------------------------------------
__global__ __launch_bounds__(256) void gat_agg_kernel(const float* __restrict__ xl,
                                                      const float* __restrict__ exw,
                                                      const float* __restrict__ denom,
                                                      const int* __restrict__ ei,
                                                      float* __restrict__ out) {
    long long gid = (long long)blockIdx.x * blockDim.x + threadIdx.x;
    int e    = (int)(gid >> 5);
    int lane = (int)(gid & 31);
    if (e >= N_EDGES) return;
    int src = ei[e];
    int dst = ei[N_EDGES + e];
    int h = lane >> 3;
    float w = exw[e * 4 + h] / (denom[dst * 4 + h] + EPS);
    float4 v = *(const float4*)(xl + (long long)src * HC + lane * 4);
    float* o = out + (long long)dst * HC + lane * 4;
    atomicAdd(o + 0, v.x * w);
    atomicAdd(o + 1, v.y * w);
    atomicAdd(o + 2, v.z * w);
    atomicAdd(o + 3, v.w * w);
}

// ---------------------------------------------------------------------------
extern "C" void kernel_launch(void* const* d_in, const int* in_sizes, int n_in,
                              void* d_out, int out_size, void* d_ws, size_t ws_size,
                              hipStream_t stream) {
    const float* x         = (const float*)d_in[0];
    const float* edge_attr = (const float*)d_in[1];
    const float* Wl        = (const float*)d_in[2];
    const float* We        = (const float*)d_in[3];
    const float* att_l     = (const float*)d_in[4];
    const float* att_r     = (const float*)d_in[5];
    const float* att_e     = (const float*)d_in[6];
    const float* bias      = (const float*)d_in[7];
    const int*   ei        = (const int*)d_in[8];
    float* out             = (float*)d_out;

    // workspace carve-up (~83 MB)
    float*    xl      = (float*)d_ws;                       // N*128
    float*    alpha   = xl + (size_t)N_NODES * HC;          // E*4
    unsigned* amaxU   = (unsigned*)(alpha + (size_t)N_EDGES * HEADS); // N*4
    float*    denom   = (float*)(amaxU + (size_t)N_NODES * HEADS);    // N*4
    float*    alpha_l = denom + (size_t)N_NODES * HEADS;    // N*4
    float*    alpha_r = alpha_l + (size_t)N_NODES * HEADS;  // N*4
    float*    ve      = alpha_r + (size_t)N_NODES * HEADS;  // 128

    gat_init_kernel<<<2048, 256, 0, stream>>>(out, bias, amaxU, denom);

    gat_gemm_xl_kernel<<<N_NODES / 16, 256, 0, stream>>>(x, Wl, xl);

    gat_alpha_lr_kernel<<<(N_NODES * HEADS + 255) / 256, 256, 0, stream>>>(
        xl, att_l, att_r, alpha_l, alpha_r);

    gat_ve_kernel<<<1, 128, 0, stream>>>(We, att_e, ve);

    gat_logit_kernel<<<(N_EDGES + 255) / 256, 256, 0, stream>>>(
        edge_attr, ve, alpha_l, alpha_r, ei, alpha, amaxU);

    gat_exp_kernel<<<(N_EDGES * HEADS + 255) / 256, 256, 0, stream>>>(
        alpha, amaxU, ei, denom);

    long long aggThreads = (long long)N_EDGES * 32;
    gat_agg_kernel<<<(unsigned)((aggThreads + 255) / 256), 256, 0, stream>>>(
        xl, alpha, denom, ei, out);
}